// GCN_APPNP_73658689126826
// MI455X (gfx1250) — compile-verified
//
#include <hip/hip_runtime.h>
#include <hip/hip_bf16.h>

// ---------------------------------------------------------------------------
// GCN (lin1+ReLU -> GCNConv -> lin2) + APPNP(K=5, alpha=0.2) + log_softmax
// MI455X (gfx1250, wave32). Dense GEMMs use V_WMMA_F32_16X16X4_F32 (exact f32).
// Sparse propagation uses hardware f32 global atomics (L2-resident working set).
// ---------------------------------------------------------------------------

typedef float v2f __attribute__((ext_vector_type(2)));
typedef float v8f __attribute__((ext_vector_type(8)));

#define HID 48
#define ALPHA_F 0.2f

static __device__ __forceinline__ void atomic_add_f32(float* p, float v) {
    (void)__hip_atomic_fetch_add(p, v, __ATOMIC_RELAXED, __HIP_MEMORY_SCOPE_AGENT);
}

// ---------------- degree / norm ----------------

__global__ void k_init_deg(float* __restrict__ deg, int n) {
    int i = blockIdx.x * blockDim.x + threadIdx.x;
    if (i < n) deg[i] = 1.0f;  // self-loop weight
}

__global__ void k_deg_edge(const int* __restrict__ col, const float* __restrict__ w,
                           float* __restrict__ deg, int E) {
    int e = blockIdx.x * blockDim.x + threadIdx.x;
    if (e < E) atomic_add_f32(&deg[col[e]], w[e]);
}

__global__ void k_rsqrt(float* __restrict__ deg, int n) {
    int i = blockIdx.x * blockDim.x + threadIdx.x;
    if (i < n) {
        float d = deg[i];
        deg[i] = (d > 0.0f) ? rsqrtf(d) : 0.0f;   // deg >= 1 always, but match ref
    }
}

__global__ void k_norm(const int* __restrict__ row, const int* __restrict__ col,
                       const float* __restrict__ w, const float* __restrict__ dinv,
                       float* __restrict__ norm, int E) {
    int e = blockIdx.x * blockDim.x + threadIdx.x;
    if (e < E) norm[e] = dinv[row[e]] * w[e] * dinv[col[e]];
}

// ---------------- dense GEMM via f32 WMMA ----------------
// out[M x 48] = act(A[M x K] @ W[K x 48] (+ bias))
// One wave handles one 16-row tile, 3 column tiles of 16 (HID == 48).
// f32 WMMA 16x16x4 VGPR layout (ISA 7.12.2):
//   A: lane = M (0..15 both halves), vgpr v holds K = (lane>=16 ? 2 : 0) + v
//   B: lane = N,                      vgpr v holds K = (lane>=16 ? 2 : 0) + v
//   C/D: lane = N, vgpr v holds M = v + (lane>=16 ? 8 : 0)
// Out-of-range A rows are clamped to M-1 (they only affect out rows that are
// never stored), so EXEC stays full and the inner loop has no divergence.
__global__ __launch_bounds__(256)
void k_gemm48(const float* __restrict__ A, const float* __restrict__ W,
              const float* __restrict__ bias, float* __restrict__ out,
              int M, int K, int do_relu) {
    int wave = threadIdx.x >> 5;
    int lane = threadIdx.x & 31;
    int tile = blockIdx.x * 8 + wave;
    int m0   = tile * 16;
    if (m0 >= M) return;                // uniform per wave

    int mr = lane & 15;                 // M (for A) / N (for B,C)
    int hi = lane >> 4;                 // 0 or 1
    int kb = hi * 2;                    // K offset of this half-wave

    int arow = m0 + mr;
    if (arow > M - 1) arow = M - 1;     // clamp: no divergence, result discarded
    const float* Arow = A + (size_t)arow * K + kb;   // 8B-aligned (K even, kb even)

    v8f c0 = {}, c1 = {}, c2 = {};

    for (int k0 = 0; k0 < K; k0 += 4) {
        v2f a = *(const v2f*)(Arow + k0);            // one global_load_b64

        const float* Wr0 = W + (size_t)(k0 + kb) * HID;
        v2f b0, b1, b2;
        b0.x = Wr0[mr];        b0.y = Wr0[HID + mr];
        b1.x = Wr0[16 + mr];   b1.y = Wr0[HID + 16 + mr];
        b2.x = Wr0[32 + mr];   b2.y = Wr0[HID + 32 + mr];

        c0 = __builtin_amdgcn_wmma_f32_16x16x4_f32(false, a, false, b0, (short)0, c0, false, false);
        c1 = __builtin_amdgcn_wmma_f32_16x16x4_f32(false, a, false, b1, (short)0, c1, false, false);
        c2 = __builtin_amdgcn_wmma_f32_16x16x4_f32(false, a, false, b2, (short)0, c2, false, false);
    }

    float bb0 = bias ? bias[mr]      : 0.0f;
    float bb1 = bias ? bias[16 + mr] : 0.0f;
    float bb2 = bias ? bias[32 + mr] : 0.0f;

    #pragma unroll
    for (int v = 0; v < 8; ++v) {
        int m = m0 + v + 8 * hi;
        if (m >= M) continue;
        float x0 = c0[v] + bb0;
        float x1 = c1[v] + bb1;
        float x2 = c2[v] + bb2;
        if (do_relu) { x0 = fmaxf(x0, 0.0f); x1 = fmaxf(x1, 0.0f); x2 = fmaxf(x2, 0.0f); }
        float* o = out + (size_t)m * HID;
        o[mr]      = x0;
        o[16 + mr] = x1;
        o[32 + mr] = x2;
    }
}

// ---------------- 48-wide propagation ----------------

// one thread per float4 group: totalVec = N*12
__global__ void k_prop48_init(const float* __restrict__ dinv, const float4* __restrict__ src,
                              float4* __restrict__ dst, int totalVec) {
    int i = blockIdx.x * blockDim.x + threadIdx.x;
    if (i < totalVec) {
        float s = dinv[i / 12];
        s = s * s;
        float4 v = src[i];
        dst[i] = make_float4(s * v.x, s * v.y, s * v.z, s * v.w);   // self-loop term
    }
}

// one thread per (edge, group of 4 features): 12 groups of float4 per edge
__global__ void k_prop48_edge(const int* __restrict__ row, const int* __restrict__ col,
                              const float* __restrict__ norm, const float* __restrict__ src,
                              float* __restrict__ dst, int E) {
    int t = blockIdx.x * blockDim.x + threadIdx.x;
    if (t >= E * 12) return;
    int e  = t / 12;
    int fg = t % 12;
    int r  = row[e];
    int c  = col[e];
    float nv = norm[e];
    const float4 v = *(const float4*)(src + (size_t)r * HID + fg * 4);
    float* d = dst + (size_t)c * HID + fg * 4;
    atomic_add_f32(d + 0, nv * v.x);
    atomic_add_f32(d + 1, nv * v.y);
    atomic_add_f32(d + 2, nv * v.z);
    atomic_add_f32(d + 3, nv * v.w);
}

// ---------------- +bc, ReLU, lin2 (48 -> 2), save h and teleport z ----------------

__global__ void k_lin2(const float* __restrict__ agg, const float* __restrict__ bc,
                       const float* __restrict__ W2, const float* __restrict__ b2,
                       float* __restrict__ h, float* __restrict__ z, int n) {
    int i = blockIdx.x * blockDim.x + threadIdx.x;
    if (i >= n) return;
    const float4* a4  = (const float4*)(agg + (size_t)i * HID);
    const float4* bc4 = (const float4*)bc;
    float d0 = b2[0], d1 = b2[1];
    #pragma unroll
    for (int g = 0; g < 12; ++g) {
        float4 av = a4[g];
        float4 bv = bc4[g];
        float v0 = fmaxf(av.x + bv.x, 0.0f);
        float v1 = fmaxf(av.y + bv.y, 0.0f);
        float v2 = fmaxf(av.z + bv.z, 0.0f);
        float v3 = fmaxf(av.w + bv.w, 0.0f);
        int f = g * 4;
        d0 = fmaf(v0, W2[(f + 0) * 2 + 0], d0);  d1 = fmaf(v0, W2[(f + 0) * 2 + 1], d1);
        d0 = fmaf(v1, W2[(f + 1) * 2 + 0], d0);  d1 = fmaf(v1, W2[(f + 1) * 2 + 1], d1);
        d0 = fmaf(v2, W2[(f + 2) * 2 + 0], d0);  d1 = fmaf(v2, W2[(f + 2) * 2 + 1], d1);
        d0 = fmaf(v3, W2[(f + 3) * 2 + 0], d0);  d1 = fmaf(v3, W2[(f + 3) * 2 + 1], d1);
    }
    h[i * 2 + 0] = d0; h[i * 2 + 1] = d1;
    z[i * 2 + 0] = d0; z[i * 2 + 1] = d1;
}

// ---------------- APPNP (2-wide propagation) ----------------

__global__ void k_appnp_self(const float* __restrict__ dinv, const float* __restrict__ h,
                             float* __restrict__ tmp, int n) {
    int i = blockIdx.x * blockDim.x + threadIdx.x;
    if (i >= n) return;
    float s = dinv[i]; s = s * s;
    tmp[i * 2 + 0] = s * h[i * 2 + 0];
    tmp[i * 2 + 1] = s * h[i * 2 + 1];
}

__global__ void k_appnp_edge(const int* __restrict__ row, const int* __restrict__ col,
                             const float* __restrict__ norm, const float* __restrict__ h,
                             float* __restrict__ tmp, int E) {
    int e = blockIdx.x * blockDim.x + threadIdx.x;
    if (e >= E) return;
    int r = row[e], c = col[e];
    float nv = norm[e];
    const float2 v = *(const float2*)(h + (size_t)r * 2);
    atomic_add_f32(tmp + (size_t)c * 2 + 0, nv * v.x);
    atomic_add_f32(tmp + (size_t)c * 2 + 1, nv * v.y);
}

__global__ void k_appnp_combine(const float* __restrict__ tmp, const float* __restrict__ z,
                                float* __restrict__ h, int n) {
    int i = blockIdx.x * blockDim.x + threadIdx.x;
    if (i >= n) return;
    h[i * 2 + 0] = (1.0f - ALPHA_F) * tmp[i * 2 + 0] + ALPHA_F * z[i * 2 + 0];
    h[i * 2 + 1] = (1.0f - ALPHA_F) * tmp[i * 2 + 1] + ALPHA_F * z[i * 2 + 1];
}

// ---------------- log_softmax over 2 classes ----------------

__global__ void k_logsoftmax2(const float* __restrict__ h, float* __restrict__ out, int n) {
    int i = blockIdx.x * blockDim.x + threadIdx.x;
    if (i >= n) return;
    float a = h[i * 2 + 0], b = h[i * 2 + 1];
    float m = fmaxf(a, b);
    float lse = m + logf(expf(a - m) + expf(b - m));
    out[i * 2 + 0] = a - lse;
    out[i * 2 + 1] = b - lse;
}

// ---------------------------------------------------------------------------

extern "C" void kernel_launch(void* const* d_in, const int* in_sizes, int n_in,
                              void* d_out, int out_size, void* d_ws, size_t ws_size,
                              hipStream_t stream) {
    (void)n_in; (void)out_size; (void)ws_size;

    const float* x  = (const float*)d_in[0];
    const int*   ei = (const int*)  d_in[1];   // [2, E]
    const float* ew = (const float*)d_in[2];
    const float* W1 = (const float*)d_in[3];
    const float* b1 = (const float*)d_in[4];
    const float* Wc = (const float*)d_in[5];
    const float* bc = (const float*)d_in[6];
    const float* W2 = (const float*)d_in[7];
    const float* b2 = (const float*)d_in[8];
    float* out = (float*)d_out;

    const int N = in_sizes[0] / 16;   // F_IN = 16
    const int E = in_sizes[2];
    const int* row = ei;
    const int* col = ei + E;

    // workspace carve-up (floats); offsets keep 16B alignment for float4 paths
    float* ws   = (float*)d_ws;
    float* dinv = ws;                 ws += N;            // degree -> dinv (in place)
    float* norm = ws;                 ws += E;
    float* bufA = ws;                 ws += (size_t)N * HID;   // h1, later agg
    float* bufB = ws;                 ws += (size_t)N * HID;   // hWc
    float* h    = ws;                 ws += (size_t)N * 2;
    float* z    = ws;                 ws += (size_t)N * 2;
    float* tmp  = ws;                 ws += (size_t)N * 2;

    const int B = 256;
    auto cdiv = [](int a, int b) { return (a + b - 1) / b; };

    // 1) symmetric normalization
    k_init_deg<<<cdiv(N, B), B, 0, stream>>>(dinv, N);
    k_deg_edge<<<cdiv(E, B), B, 0, stream>>>(col, ew, dinv, E);
    k_rsqrt  <<<cdiv(N, B), B, 0, stream>>>(dinv, N);
    k_norm   <<<cdiv(E, B), B, 0, stream>>>(row, col, ew, dinv, norm, E);

    // 2) h1 = relu(x @ W1 + b1)      [WMMA f32, K=16]
    int tiles = cdiv(N, 16);
    k_gemm48<<<cdiv(tiles, 8), 256, 0, stream>>>(x, W1, b1, bufA, N, 16, 1);

    // 3) hWc = h1 @ Wc               [WMMA f32, K=48]
    k_gemm48<<<cdiv(tiles, 8), 256, 0, stream>>>(bufA, Wc, nullptr, bufB, N, HID, 0);

    // 4) agg = D^-1/2 A D^-1/2 @ hWc   (bufB -> bufA)
    k_prop48_init<<<cdiv(N * 12, B), B, 0, stream>>>(dinv, (const float4*)bufB,
                                                     (float4*)bufA, N * 12);
    k_prop48_edge<<<cdiv(E * 12, B), B, 0, stream>>>(row, col, norm, bufB, bufA, E);

    // 5) h = z = relu(agg + bc) @ W2 + b2
    k_lin2<<<cdiv(N, B), B, 0, stream>>>(bufA, bc, W2, b2, h, z, N);

    // 6) APPNP: 5 power-iteration steps
    for (int it = 0; it < 5; ++it) {
        k_appnp_self   <<<cdiv(N, B), B, 0, stream>>>(dinv, h, tmp, N);
        k_appnp_edge   <<<cdiv(E, B), B, 0, stream>>>(row, col, norm, h, tmp, E);
        k_appnp_combine<<<cdiv(N, B), B, 0, stream>>>(tmp, z, h, N);
    }

    // 7) log_softmax
    k_logsoftmax2<<<cdiv(N, B), B, 0, stream>>>(h, out, N);
}